// RZRZZ_9354438771405
// MI455X (gfx1250) — compile-verified
//
#include <hip/hip_runtime.h>

typedef __attribute__((ext_vector_type(2))) float v2f;
typedef __attribute__((ext_vector_type(8))) float v8f;

// ---------------------------------------------------------------------------
// Kernel 1: zero-fill the 128 MiB output with wide b128 stores (bandwidth
// bound part of the problem; 23.3 TB/s HBM -> ~5.5us floor).
// ---------------------------------------------------------------------------
__global__ __launch_bounds__(256) void rzrzz_zero_fill(float4* __restrict__ out,
                                                       long long n4) {
    long long i = (long long)blockIdx.x * blockDim.x + threadIdx.x;
    const long long stride = (long long)gridDim.x * blockDim.x;
    const float4 z = make_float4(0.f, 0.f, 0.f, 0.f);
    for (; i < n4; i += stride) {
        out[i] = z;
    }
}

// ---------------------------------------------------------------------------
// Branch-free sincos: Cody-Waite 3-step pi/2 reduction + minimax polynomials.
// Accurate to ~1-2 ulp for |x| < ~1e4; theta here is O(10) at most.
// Avoids libm's Payne-Hanek big-argument block (64-bit mads + exec branches).
// ---------------------------------------------------------------------------
__device__ __forceinline__ void sincos_fast(float x, float* sp, float* cp) {
    const float twoOverPi = 0.636619772367581343f;
    const float pio2_hi   = 1.57079601287841796875f;   // 24-bit chunk
    const float pio2_mid  = 3.1391647326017846353e-7f;
    const float pio2_lo   = 5.3903025299577647655e-15f;

    float fn = rintf(x * twoOverPi);           // v_rndne_f32
    int   q  = (int)fn;
    float r  = fmaf(-fn, pio2_hi,  x);
    r        = fmaf(-fn, pio2_mid, r);
    r        = fmaf(-fn, pio2_lo,  r);
    float r2 = r * r;

    // sin(r) ~= r + r^3 * P(r^2), cos(r) ~= 1 + r^2 * Q(r^2) on [-pi/4, pi/4]
    float ps = fmaf(r2, fmaf(r2, -1.9840874e-4f, 8.3333310e-3f), -1.6666667e-1f);
    ps       = fmaf(r * r2, ps, r);
    float pc = fmaf(r2, fmaf(r2, -1.3887316e-3f, 4.1666646e-2f), -0.5f);
    pc       = fmaf(r2, pc, 1.0f);

    // quadrant: sin(x) = {ps, pc, -ps, -pc}[q&3]; cos(x) = {pc, -ps, -pc, ps}
    bool swap = (q & 1) != 0;
    float s = swap ? pc : ps;
    float c = swap ? ps : pc;
    unsigned ssign = ((unsigned)q & 2u) << 30;         // negate if q%4 in {2,3}
    unsigned csign = (((unsigned)q + 1u) & 2u) << 30;  // negate if q%4 in {1,2}
    *sp = __uint_as_float(__float_as_uint(s) ^ ssign);
    *cp = __uint_as_float(__float_as_uint(c) ^ csign);
}

// ---------------------------------------------------------------------------
// Kernel 2: per wave, compute theta for a 16-batch tile with
// V_WMMA_F32_16X16X4_F32 (K=12, padded from 10), then scatter the complex
// diagonal values exp(i*theta) with aligned float2 stores.
//
// theta = [x0x1 x0x2 x0x3 x1x2 x1x3 x2x3 x0 x1 x2 x3 0 0] @ W
// W[p][k]   = 0.5  * rzz_sign[p*16+k]       (p < 6)
// W[6+q][k] = -0.5 * rz_static[k*4+q]       (q < 4)
// W[10..11] = 0
//
// All lane-dependent operand construction uses named scalars + ternaries so
// it lowers to v_cndmask (no LDS spills, no EXEC-masked branches).
// ---------------------------------------------------------------------------
__global__ __launch_bounds__(256) void rzrzz_theta_diag(
        const float* __restrict__ x,
        const float* __restrict__ rzz_sign,
        const float* __restrict__ rz_static,
        float* __restrict__ out) {
    const int lane = threadIdx.x & 31;
    const int wave = threadIdx.x >> 5;
    const int tile = blockIdx.x * 8 + wave;   // 8 waves/block, 16 rows/tile
    const int b0   = tile * 16;

    const bool hi  = (lane & 16) != 0;  // lanes 16-31 carry K = {2,3} slices
    const int  l16 = lane & 15;

    // ---- A-matrix row for this lane: M = l16 (same row for both halves) ---
    const float4 xr = *(const float4*)(x + (size_t)(b0 + l16) * 4);
    const float p01 = xr.x * xr.y;
    const float p02 = xr.x * xr.z;
    const float p03 = xr.x * xr.w;
    const float p12 = xr.y * xr.z;
    const float p13 = xr.y * xr.w;
    const float p23 = xr.z * xr.w;

    // ---- W column l16, all rows this lane might touch (unconditional) -----
    const float w0 = 0.5f * rzz_sign[0 * 16 + l16];
    const float w1 = 0.5f * rzz_sign[1 * 16 + l16];
    const float w2 = 0.5f * rzz_sign[2 * 16 + l16];
    const float w3 = 0.5f * rzz_sign[3 * 16 + l16];
    const float w4 = 0.5f * rzz_sign[4 * 16 + l16];
    const float w5 = 0.5f * rzz_sign[5 * 16 + l16];
    const float4 zr = *(const float4*)(rz_static + (size_t)l16 * 4);
    const float w6 = -0.5f * zr.x;
    const float w7 = -0.5f * zr.y;
    const float w8 = -0.5f * zr.z;
    const float w9 = -0.5f * zr.w;

    v8f c = {};
    v2f a, b;

    // j = 0: K = 0..3  -> A cols {p01,p02 | p03,p12}, W rows {0,1 | 2,3}
    a.x = hi ? p03 : p01;
    a.y = hi ? p12 : p02;
    b.x = hi ? w2 : w0;
    b.y = hi ? w3 : w1;
    c = __builtin_amdgcn_wmma_f32_16x16x4_f32(false, a, false, b, (short)0, c,
                                              false, false);

    // j = 1: K = 4..7  -> A cols {p13,p23 | x0,x1}, W rows {4,5 | 6,7}
    a.x = hi ? xr.x : p13;
    a.y = hi ? xr.y : p23;
    b.x = hi ? w6 : w4;
    b.y = hi ? w7 : w5;
    c = __builtin_amdgcn_wmma_f32_16x16x4_f32(false, a, false, b, (short)0, c,
                                              false, false);

    // j = 2: K = 8..11 -> A cols {x2,x3 | 0,0}, W rows {8,9 | zero pad}
    a.x = hi ? 0.0f : xr.z;
    a.y = hi ? 0.0f : xr.w;
    b.x = hi ? 0.0f : w8;
    b.y = hi ? 0.0f : w9;
    c = __builtin_amdgcn_wmma_f32_16x16x4_f32(false, a, false, b, (short)0, c,
                                              false, false);

    // D layout: lane, VGPR v -> (M = v + 8*hi, N = l16).
    // out[b, k, k] at flat float offset b*512 + k*34 (k*17 float2s, aligned).
    const int browBase = b0 + (hi ? 8 : 0);
#pragma unroll
    for (int v = 0; v < 8; ++v) {
        float s, co;
        sincos_fast(c[v], &s, &co);
        float2 val = make_float2(co, s);  // real = cos, imag = sin
        *(float2*)(out + (size_t)(browBase + v) * 512 + (size_t)l16 * 34) = val;
    }
}

extern "C" void kernel_launch(void* const* d_in, const int* in_sizes, int n_in,
                              void* d_out, int out_size, void* d_ws, size_t ws_size,
                              hipStream_t stream) {
    (void)n_in; (void)out_size; (void)d_ws; (void)ws_size;
    const float* x         = (const float*)d_in[0];   // (B, 4) f32
    const float* rzz_sign  = (const float*)d_in[1];   // (6, 16) f32
    const float* rz_static = (const float*)d_in[2];   // (16, 4) f32
    // d_in[3] = pairs, order hardcoded above.
    float* out = (float*)d_out;                       // (B, 16, 16) complex64

    const int batch = in_sizes[0] / 4;                // 65536
    const long long n4 = (long long)batch * 128;      // float4 count of output

    rzrzz_zero_fill<<<2048, 256, 0, stream>>>((float4*)out, n4);

    const int blocks = batch / 128;                   // 8 waves * 16 rows each
    rzrzz_theta_diag<<<blocks, 256, 0, stream>>>(x, rzz_sign, rz_static, out);
}